// SteinerTopo_30236569763932
// MI455X (gfx1250) — compile-verified
//
#include <hip/hip_runtime.h>

// Pure double-indirection gather:
//   out_x[v] = pos[ l2g[prx[v]] ]
//   out_y[v] = pos[ P + l2g[pry[v]] ]
// Memory-bound: ~304 MB min traffic -> ~13 us at 23.3 TB/s.
// Strategy: NT loads/stores for the streamed index/output arrays so the
// randomly-gathered tables (l2g 48MB + pos 64MB) stay resident in the 192MB L2;
// gathers use default RT policy and hit L2 after first touch.

typedef int   v4i __attribute__((ext_vector_type(4)));
typedef float v4f __attribute__((ext_vector_type(4)));

__global__ __launch_bounds__(256) void steiner_topo_gather(
    const float* __restrict__ pos_x,   // pos[0 .. P)
    const float* __restrict__ pos_y,   // pos[P .. 2P)
    const int*   __restrict__ prx,     // [V]
    const int*   __restrict__ pry,     // [V]
    const int*   __restrict__ l2g,     // [V] -> pin ids in [0, P)
    float*       __restrict__ out_x,   // [V]
    float*       __restrict__ out_y,   // [V]
    int          num_vertices)
{
    const int i4   = blockIdx.x * blockDim.x + threadIdx.x;
    const int base = i4 << 2;

    if (base + 3 < num_vertices) {
        // Streamed, read-once index vectors: non-temporal B128 loads
        // (don't pollute L2, which we want to keep for the gather tables).
        v4i rx = __builtin_nontemporal_load((const v4i*)prx + i4);
        v4i ry = __builtin_nontemporal_load((const v4i*)pry + i4);

        // First indirection: 8 independent RT gathers into l2g (L2-resident).
        int gx0 = l2g[rx.x], gx1 = l2g[rx.y], gx2 = l2g[rx.z], gx3 = l2g[rx.w];
        int gy0 = l2g[ry.x], gy1 = l2g[ry.y], gy2 = l2g[ry.z], gy3 = l2g[ry.w];

        // Second indirection: 8 independent RT gathers into pos (L2-resident).
        v4f ox, oy;
        ox.x = pos_x[gx0]; ox.y = pos_x[gx1]; ox.z = pos_x[gx2]; ox.w = pos_x[gx3];
        oy.x = pos_y[gy0]; oy.y = pos_y[gy1]; oy.z = pos_y[gy2]; oy.w = pos_y[gy3];

        // Streamed, write-once outputs: non-temporal B128 stores.
        __builtin_nontemporal_store(ox, (v4f*)out_x + i4);
        __builtin_nontemporal_store(oy, (v4f*)out_y + i4);
    } else if (base < num_vertices) {
        // Tail (never taken for V = 12,000,000, kept for safety).
        for (int v = base; v < num_vertices; ++v) {
            out_x[v] = pos_x[l2g[prx[v]]];
            out_y[v] = pos_y[l2g[pry[v]]];
        }
    }
}

extern "C" void kernel_launch(void* const* d_in, const int* in_sizes, int n_in,
                              void* d_out, int out_size, void* d_ws, size_t ws_size,
                              hipStream_t stream) {
    // Inputs in setup_inputs() order:
    //   0: pos               [2P]  float32
    //   1: pin_relate_x      [V]   int32
    //   2: pin_relate_y      [V]   int32
    //   3: local2global_idx  [V]   int32
    //   4: net_vertex_start  [NETS+1] int32   (unused by the math)
    //   5: num_total_vertices scalar         (use in_sizes[1] instead)
    const float* pos = (const float*)d_in[0];
    const int*   prx = (const int*)d_in[1];
    const int*   pry = (const int*)d_in[2];
    const int*   l2g = (const int*)d_in[3];

    const int P = in_sizes[0] / 2;   // number of pins
    const int V = in_sizes[1];       // number of vertices

    float* out_x = (float*)d_out;        // tuple element 0
    float* out_y = (float*)d_out + V;    // tuple element 1

    const int n_vec4   = (V + 3) / 4;            // threads (4 vertices each)
    const int threads  = 256;                    // 8 wave32 waves per block
    const int blocks   = (n_vec4 + threads - 1) / threads;

    steiner_topo_gather<<<blocks, threads, 0, stream>>>(
        pos, pos + P, prx, pry, l2g, out_x, out_y, V);
}